// Seq2Seq_79877801771156
// MI455X (gfx1250) — compile-verified
//
#include <hip/hip_runtime.h>
#include <hip/hip_bf16.h>

// Seq2Seq GRU for MI455X (gfx1250), fp32 end-to-end via V_WMMA_F32_16X16X4_F32.
// - Each wave32 privately owns 16 batch rows; h stays in wave-private LDS for
//   all 384 recurrent steps.
// - Weights staged in LDS in WMMA-operand order (one float2 per
//   (Ntile,Kstep,lane) slot): each B operand is one conflict-free ds_load_b64.
// - Single dynamic-LDS blob => compiler cannot hoist the weight loads into
//   registers across the time loop (fixes round-1 scratch-spill serialization).

typedef __attribute__((ext_vector_type(2))) float v2f;
typedef __attribute__((ext_vector_type(8))) float v8f;

#define SEQ_LEN   256
#define LABEL_LEN 128
#define BATCH     8192
#define NFEAT     8
#define NOUT      8
#define HID       64

#define BLOCK_THREADS 128
#define WAVES     4
#define RPW       16                 // rows per wave
#define ROWS      (RPW * WAVES)      // 64 rows per block
#define H_S       68                 // LDS row stride (floats) for h
#define INP_S     10                 // LDS row stride for decoder input staging

// ---- dynamic LDS partition (float offsets) ----
#define SM_WHH   0                         // 12*16*32 float2 = 12288 floats
#define SM_WIH   (SM_WHH + 12288)          // 12*2*32 float2  = 1536 floats
#define SM_WOUT  (SM_WIH + 1536)           // 16*32 float2    = 1024 floats
#define SM_BIH   (SM_WOUT + 1024)          // 192
#define SM_BHH   (SM_BIH + 192)            // 192
#define SM_H     (SM_BHH + 192)            // ROWS*H_S = 4352
#define SM_INP   (SM_H + ROWS * H_S)       // ROWS*INP_S = 640
#define SM_TOTAL (SM_INP + ROWS * INP_S)   // 20224 floats = 80896 B

// D(16x16 f32) = A(16x4 f32) * B(4x16 f32) + C
static __device__ __forceinline__ v8f wmma4(v2f a, v2f b, v8f c) {
  return __builtin_amdgcn_wmma_f32_16x16x4_f32(false, a, false, b, (short)0, c,
                                               false, false);
}
static __device__ __forceinline__ float sigmoidf_(float x) {
  return 1.0f / (1.0f + __expf(-x));
}
static __device__ __forceinline__ float tanhf_(float x) {
  float e = __expf(2.0f * x);
  return (e - 1.0f) / (e + 1.0f);
}

// Stage GRU weights into LDS in WMMA-operand order.
// B operand for lane l, vgpr v of tile (nt,ks):  W[nt*16 + (l&15)][ks*4 + 2*(l>>4) + v]
static __device__ void stage_gru_weights(float* __restrict__ smem,
                                         const float* __restrict__ Wih,
                                         const float* __restrict__ Whh,
                                         const float* __restrict__ bih,
                                         const float* __restrict__ bhh,
                                         int tid) {
  v2f* whhB = (v2f*)(smem + SM_WHH);
  for (int s = tid; s < 12 * 16 * 32; s += BLOCK_THREADS) {
    int nt = s >> 9, ks = (s >> 5) & 15, l = s & 31;
    int n = nt * 16 + (l & 15);
    int k = ks * 4 + 2 * (l >> 4);
    v2f v; v.x = Whh[n * HID + k]; v.y = Whh[n * HID + k + 1];
    whhB[s] = v;
  }
  v2f* wihB = (v2f*)(smem + SM_WIH);
  for (int s = tid; s < 12 * 2 * 32; s += BLOCK_THREADS) {
    int nt = s >> 6, ks = (s >> 5) & 1, l = s & 31;
    int n = nt * 16 + (l & 15);
    int k = ks * 4 + 2 * (l >> 4);
    v2f v; v.x = Wih[n * NFEAT + k]; v.y = Wih[n * NFEAT + k + 1];
    wihB[s] = v;
  }
  for (int i = tid; i < 192; i += BLOCK_THREADS) {
    smem[SM_BIH + i] = bih[i];
    smem[SM_BHH + i] = bhh[i];
  }
}

// One GRU cell update for this wave's 16 rows (PyTorch gate order r,z,n).
static __device__ __forceinline__ void gru_step(float* __restrict__ smem,
                                                float* __restrict__ hRow,
                                                v2f ai0, v2f ai1,
                                                int lane, int li, int lh) {
  const v2f* whhB = (const v2f*)(smem + SM_WHH);
  const v2f* wihB = (const v2f*)(smem + SM_WIH);

  // Preload A operands for h (reused by all 12 N-tiles).
  v2f ah[16];
#pragma unroll
  for (int ks = 0; ks < 16; ++ks)
    ah[ks] = *(const v2f*)&hRow[li * H_S + ks * 4 + 2 * lh];

  v8f r[4], z[4];
  // r tiles (nt 0..3) and z tiles (nt 4..7): gi and gh share accumulation.
#pragma unroll
  for (int j = 0; j < 8; ++j) {
    const int nt = j;
    v8f a0 = {}, a1 = {};        // two independent wmma chains
#pragma unroll
    for (int ks = 0; ks < 16; ks += 2) {
      a0 = wmma4(ah[ks],     whhB[(nt * 16 + ks) * 32 + lane],     a0);
      a1 = wmma4(ah[ks + 1], whhB[(nt * 16 + ks + 1) * 32 + lane], a1);
    }
    a0 = wmma4(ai0, wihB[(nt * 2 + 0) * 32 + lane], a0);
    a1 = wmma4(ai1, wihB[(nt * 2 + 1) * 32 + lane], a1);
    const float bsum = smem[SM_BIH + nt * 16 + li] + smem[SM_BHH + nt * 16 + li];
    v8f gate;
#pragma unroll
    for (int e = 0; e < 8; ++e) gate[e] = sigmoidf_(a0[e] + a1[e] + bsum);
    if (j < 4) r[j] = gate; else z[j - 4] = gate;
  }

  // n tiles (nt 8..11): n = tanh(inn + bi + r*(hn + bh))
#pragma unroll
  for (int j = 0; j < 4; ++j) {
    const int nt = j + 8;
    v8f h0 = {}, h1 = {};
#pragma unroll
    for (int ks = 0; ks < 16; ks += 2) {
      h0 = wmma4(ah[ks],     whhB[(nt * 16 + ks) * 32 + lane],     h0);
      h1 = wmma4(ah[ks + 1], whhB[(nt * 16 + ks + 1) * 32 + lane], h1);
    }
    v8f gi = {};
    gi = wmma4(ai0, wihB[(nt * 2 + 0) * 32 + lane], gi);
    gi = wmma4(ai1, wihB[(nt * 2 + 1) * 32 + lane], gi);
    const float bi = smem[SM_BIH + nt * 16 + li];
    const float bh = smem[SM_BHH + nt * 16 + li];
    // D layout: row = e + 8*lh, col = j*16 + li (matches r[j], z[j], h cols)
#pragma unroll
    for (int e = 0; e < 8; ++e) {
      float n = tanhf_(gi[e] + bi + r[j][e] * (h0[e] + h1[e] + bh));
      float* hp = &hRow[(e + 8 * lh) * H_S + j * 16 + li];
      float ho = *hp;
      float zz = z[j][e];
      *hp = (1.0f - zz) * n + zz * ho;   // ah[] already in regs: safe
    }
  }
}

__global__ __launch_bounds__(BLOCK_THREADS) void seq2seq_gru_kernel(
    const float* __restrict__ x,    const float* __restrict__ xy,
    const float* __restrict__ eWih, const float* __restrict__ eWhh,
    const float* __restrict__ ebih, const float* __restrict__ ebhh,
    const float* __restrict__ dWih, const float* __restrict__ dWhh,
    const float* __restrict__ dbih, const float* __restrict__ dbhh,
    const float* __restrict__ Wout, const float* __restrict__ bout,
    float* __restrict__ out)
{
  extern __shared__ __align__(16) float smem[];

  const int tid  = (int)threadIdx.x;
  const int lane = tid & 31;
  const int wave = tid >> 5;
  const int li = lane & 15;
  const int lh = lane >> 4;
  float* hRow = smem + SM_H + wave * RPW * H_S;
  float* iRow = smem + SM_INP + wave * RPW * INP_S;
  const int rowbase = (int)blockIdx.x * ROWS + wave * RPW;

  // h0 = 0 (wave-private region)
  for (int i = lane; i < RPW * H_S; i += 32) hRow[i] = 0.0f;

  // ---------------- Encoder ----------------
  stage_gru_weights(smem, eWih, eWhh, ebih, ebhh, tid);
  __syncthreads();

  for (int t = 0; t < SEQ_LEN; ++t) {
    const float* xt = x + (size_t)t * BATCH * NFEAT + (size_t)(rowbase + li) * NFEAT;
    v2f ai0 = *(const v2f*)(xt + 0 + 2 * lh);
    v2f ai1 = *(const v2f*)(xt + 4 + 2 * lh);
    gru_step(smem, hRow, ai0, ai1, lane, li, lh);
  }

  // ---------------- Decoder ----------------
  __syncthreads();   // all waves done with encoder weights
  stage_gru_weights(smem, dWih, dWhh, dbih, dbhh, tid);
  {
    v2f* woutB = (v2f*)(smem + SM_WOUT);
    for (int s = tid; s < 16 * 32; s += BLOCK_THREADS) {
      int ks = s >> 5, l = s & 31;
      int wli = l & 15, wlh = l >> 4;
      int k = ks * 4 + 2 * wlh;
      v2f v;
      if (wli < NOUT) { v.x = Wout[wli * HID + k]; v.y = Wout[wli * HID + k + 1]; }
      else            { v.x = 0.0f; v.y = 0.0f; }
      woutB[s] = v;
    }
  }
  __syncthreads();

  // inp = xy[0] for this wave's rows
  for (int i = lane; i < RPW * NOUT; i += 32) {
    int rr = i >> 3, cc = i & 7;
    iRow[rr * INP_S + cc] = xy[(size_t)(rowbase + rr) * NOUT + cc];
  }

  const v2f* woutB = (const v2f*)(smem + SM_WOUT);
  const float bb = (li < NOUT) ? bout[li] : 0.0f;

  for (int t = 0; t < LABEL_LEN; ++t) {
    v2f ai0 = *(const v2f*)&iRow[li * INP_S + 0 + 2 * lh];
    v2f ai1 = *(const v2f*)&iRow[li * INP_S + 4 + 2 * lh];
    gru_step(smem, hRow, ai0, ai1, lane, li, lh);

    // out = h @ Wout.T + bout  (M=16, K=64; N-tile cols 8..15 are zero)
    v8f o0 = {}, o1 = {};
#pragma unroll
    for (int ks = 0; ks < 16; ks += 2) {
      v2f a0 = *(const v2f*)&hRow[li * H_S + ks * 4 + 2 * lh];
      v2f a1 = *(const v2f*)&hRow[li * H_S + (ks + 1) * 4 + 2 * lh];
      o0 = wmma4(a0, woutB[ks * 32 + lane],       o0);
      o1 = wmma4(a1, woutB[(ks + 1) * 32 + lane], o1);
    }
    float* op = out + ((size_t)t * BATCH + rowbase) * NOUT;
#pragma unroll
    for (int e = 0; e < 8; ++e) {
      float val = o0[e] + o1[e] + bb;
      int rr = e + 8 * lh;
      if (li < NOUT) {
        op[(size_t)rr * NOUT + li] = val;   // result
        iRow[rr * INP_S + li] = val;        // feedback input for step t+1
      }
    }
  }
}

extern "C" void kernel_launch(void* const* d_in, const int* in_sizes, int n_in,
                              void* d_out, int out_size, void* d_ws, size_t ws_size,
                              hipStream_t stream) {
  const float* x    = (const float*)d_in[0];
  const float* xy   = (const float*)d_in[1];
  const float* eWih = (const float*)d_in[2];
  const float* eWhh = (const float*)d_in[3];
  const float* ebih = (const float*)d_in[4];
  const float* ebhh = (const float*)d_in[5];
  const float* dWih = (const float*)d_in[6];
  const float* dWhh = (const float*)d_in[7];
  const float* dbih = (const float*)d_in[8];
  const float* dbhh = (const float*)d_in[9];
  const float* Wout = (const float*)d_in[10];
  const float* bout = (const float*)d_in[11];
  float* out = (float*)d_out;
  (void)in_sizes; (void)n_in; (void)out_size; (void)d_ws; (void)ws_size;

  const size_t smem_bytes = (size_t)SM_TOTAL * sizeof(float);  // ~79 KB (<320 KB/WG on gfx1250)
  (void)hipFuncSetAttribute((const void*)seq2seq_gru_kernel,
                            hipFuncAttributeMaxDynamicSharedMemorySize,
                            (int)smem_bytes);

  dim3 grid(BATCH / ROWS);        // 128 blocks
  dim3 block(BLOCK_THREADS);      // 4 waves, 16 rows each
  seq2seq_gru_kernel<<<grid, block, smem_bytes, stream>>>(
      x, xy, eWih, eWhh, ebih, ebhh, dWih, dWhh, dbih, dbhh, Wout, bout, out);
}